// MultiHeadGatedAttention_33097017983483
// MI455X (gfx1250) — compile-verified
//
#include <hip/hip_runtime.h>
#include <math.h>

typedef __attribute__((ext_vector_type(2))) float v2f;
typedef __attribute__((ext_vector_type(8))) float v8f;

constexpr int B = 32, S = 4096, D = 1024, H = 8;
constexpr int HD = H * D; // 8192

__device__ __forceinline__ v8f wmma_f32(v2f a, v2f b, v8f c) {
  // D = A(16x4 f32) * B(4x16 f32) + C(16x16 f32)
  return __builtin_amdgcn_wmma_f32_16x16x4_f32(
      /*neg_a=*/false, a, /*neg_b=*/false, b,
      /*c_mod=*/(short)0, c, /*reuse_a=*/false, /*reuse_b=*/false);
}

// Single v_rcp_f32 (~1 ulp) instead of the IEEE divide expansion.
__device__ __forceinline__ float fast_rcp(float x) {
#if __has_builtin(__builtin_amdgcn_rcpf)
  return __builtin_amdgcn_rcpf(x);
#else
  return 1.0f / x;
#endif
}

// ---------------------------------------------------------------------------
// Kernel 1: fused score+gate GEMM.  out_tile[16 s][16 n] where n<8 = energy
// head n, n>=8 = gate pre-activation head n-8.  Gated energies written as
// [B,H,S] (the transpose in the reference).
// ---------------------------------------------------------------------------
__global__ __launch_bounds__(256) void k_score_gate(
    const float* __restrict__ x, const float* __restrict__ w_score,
    const float* __restrict__ w_gate, const float* __restrict__ b_gate,
    float* __restrict__ eg) {
  __shared__ float Wl[D * 16];  // [k][n] layout, 64 KB, conflict-free ds reads
  const int t = threadIdx.x;
  for (int n = 0; n < 16; ++n) {
    const float* src = (n < 8) ? (w_score + n * D) : (w_gate + (n - 8) * D);
    for (int k = t; k < D; k += 256) Wl[k * 16 + n] = src[k];  // coalesced read
  }
  __syncthreads();

  const int wid = t >> 5, lane = t & 31;
  const int half = lane >> 4, m = lane & 15;
  const long r0 = (long)(blockIdx.x * 8 + wid) * 16;  // 16 rows of [B*S, D]
  const int bb = (int)(r0 / S), s0 = (int)(r0 % S);
  const float* rowp = x + (r0 + m) * (long)D;

  v8f c = {};
#pragma unroll 4
  for (int k0 = 0; k0 < D; k0 += 4) {
    const int kk = k0 + 2 * half;
    v2f a = *reinterpret_cast<const v2f*>(rowp + kk);  // A: row m, K=kk,kk+1
    v2f bf;
    bf[0] = Wl[kk * 16 + m];        // B: K=kk,   N=m
    bf[1] = Wl[(kk + 1) * 16 + m];  // B: K=kk+1, N=m
    c = wmma_f32(a, bf, c);
  }

  const float bg = (m < 8) ? b_gate[m] : 0.0f;
#pragma unroll
  for (int j = 0; j < 8; ++j) {
    float e = c[j];                      // element (s0 + j + 8*half, N=m)
    float gp = __shfl_xor(e, 8, 32);     // col m+8 holds gate pre-act of head m
    if (m < 8) {
      float g = fast_rcp(1.0f + __expf(-(gp + bg)));  // sigmoid
      int s = s0 + j + 8 * half;
      eg[((long)bb * H + m) * S + s] = e * g;
    }
  }
}

// ---------------------------------------------------------------------------
// Kernel 2: softmax over S per (b,h).  One 256-thread block per row.
// ---------------------------------------------------------------------------
__device__ __forceinline__ float wred_max(float v) {
#pragma unroll
  for (int o = 16; o > 0; o >>= 1) v = fmaxf(v, __shfl_down(v, o, 32));
  return v;
}
__device__ __forceinline__ float wred_sum(float v) {
#pragma unroll
  for (int o = 16; o > 0; o >>= 1) v += __shfl_down(v, o, 32);
  return v;
}

__global__ __launch_bounds__(256) void k_softmax(const float* __restrict__ eg,
                                                 float* __restrict__ attn) {
  __shared__ float red[8];
  __shared__ float bcast;
  const long row = blockIdx.x;  // b*H + h
  const float* p = eg + row * S;
  float* o = attn + row * S;
  const int t = threadIdx.x, lane = t & 31, wid = t >> 5;

  float v[16];
  float mx = -3.4e38f;
#pragma unroll
  for (int i = 0; i < 16; ++i) {
    v[i] = p[t + i * 256];
    mx = fmaxf(mx, v[i]);
  }
  mx = wred_max(mx);
  if (lane == 0) red[wid] = mx;
  __syncthreads();
  if (wid == 0) {
    float r = (lane < 8) ? red[lane] : -3.4e38f;
    r = wred_max(r);
    if (lane == 0) bcast = r;
  }
  __syncthreads();
  mx = bcast;

  float sum = 0.0f;
#pragma unroll
  for (int i = 0; i < 16; ++i) {
    v[i] = __expf(v[i] - mx);
    sum += v[i];
  }
  sum = wred_sum(sum);
  __syncthreads();
  if (lane == 0) red[wid] = sum;
  __syncthreads();
  if (wid == 0) {
    float r = (lane < 8) ? red[lane] : 0.0f;
    r = wred_sum(r);
    if (lane == 0) bcast = r;
  }
  __syncthreads();
  const float inv = fast_rcp(bcast);
#pragma unroll
  for (int i = 0; i < 16; ++i) o[t + i * 256] = v[i] * inv;
}

// ---------------------------------------------------------------------------
// Kernel 3: head_ctx[b,h,d] = sum_s attn[b,h,s] * x[b,s,d].
// One wave per (batch, 16-column D tile): 2048 waves, x read exactly once.
// ---------------------------------------------------------------------------
__global__ __launch_bounds__(128) void k_headctx(const float* __restrict__ x,
                                                 const float* __restrict__ attn,
                                                 float* __restrict__ hc) {
  const int t = threadIdx.x, wid = t >> 5, lane = t & 31;
  const int half = lane >> 4, m = lane & 15;
  const int bb = blockIdx.x >> 4;
  const int n0 = ((blockIdx.x & 15) * 4 + wid) * 16;
  const float* xb = x + (long)bb * S * D;
  const float* ab = attn + (long)bb * H * S;

  v8f c = {};
#pragma unroll 4
  for (int k0 = 0; k0 < S; k0 += 4) {
    const int kk = k0 + 2 * half;
    v2f a = {};
    if (m < H) a = *reinterpret_cast<const v2f*>(ab + (long)m * S + kk);
    v2f bf;
    bf[0] = xb[(long)kk * D + n0 + m];        // B: K=kk,   N=m (coalesced 64B)
    bf[1] = xb[(long)(kk + 1) * D + n0 + m];  // B: K=kk+1, N=m
    c = wmma_f32(a, bf, c);
  }
  if (half == 0) {  // rows M=0..7 = valid heads
#pragma unroll
    for (int j = 0; j < 8; ++j)
      hc[((long)bb * H + j) * D + n0 + m] = c[j];
  }
}

// ---------------------------------------------------------------------------
// Kernel 4: context[b,o] = head_ctx_flat[b,:] . w_out[o,:] + b_out[o].
// One wave per 16 output cols; both 16-row batch tiles share the B fragment.
// ---------------------------------------------------------------------------
__global__ __launch_bounds__(32) void k_outproj(const float* __restrict__ hc,
                                                const float* __restrict__ w_out,
                                                const float* __restrict__ b_out,
                                                float* __restrict__ ctx) {
  const int lane = threadIdx.x & 31;
  const int half = lane >> 4, m = lane & 15;
  const int o0 = blockIdx.x * 16;
  const float* wr = w_out + (long)(o0 + m) * HD;

  v8f c0 = {}, c1 = {};
#pragma unroll 4
  for (int k0 = 0; k0 < HD; k0 += 4) {
    const int kk = k0 + 2 * half;
    v2f a0 = *reinterpret_cast<const v2f*>(hc + (long)m * HD + kk);
    v2f a1 = *reinterpret_cast<const v2f*>(hc + (long)(16 + m) * HD + kk);
    v2f bf = *reinterpret_cast<const v2f*>(wr + kk);
    c0 = wmma_f32(a0, bf, c0);
    c1 = wmma_f32(a1, bf, c1);
  }
  const float bo = b_out[o0 + m];
#pragma unroll
  for (int j = 0; j < 8; ++j) {
    ctx[(long)(j + 8 * half) * D + o0 + m] = c0[j] + bo;
    ctx[(long)(16 + j + 8 * half) * D + o0 + m] = c1[j] + bo;
  }
}

// ---------------------------------------------------------------------------
extern "C" void kernel_launch(void* const* d_in, const int* in_sizes, int n_in,
                              void* d_out, int out_size, void* d_ws,
                              size_t ws_size, hipStream_t stream) {
  const float* x       = (const float*)d_in[0];  // [B,S,D]
  const float* w_score = (const float*)d_in[1];  // [H,D]
  const float* w_gate  = (const float*)d_in[2];  // [H,D]
  const float* b_gate  = (const float*)d_in[3];  // [H]
  const float* w_out   = (const float*)d_in[4];  // [D, H*D]
  const float* b_out   = (const float*)d_in[5];  // [D]

  float* ctx  = (float*)d_out;               // [B,D]
  float* attn = ctx + (long)B * D;           // [B,H,S]
  float* eg   = (float*)d_ws;                // [B,H,S] gated energies (4 MB)
  float* hc   = eg + (long)B * H * S;        // [B,H,D] head context (1 MB)

  k_score_gate<<<(B * S) / (16 * 8), 256, 0, stream>>>(x, w_score, w_gate,
                                                       b_gate, eg);
  k_softmax<<<B * H, 256, 0, stream>>>(eg, attn);
  k_headctx<<<B * (D / 16) / 4, 128, 0, stream>>>(x, attn, hc);
  k_outproj<<<D / 16, 32, 0, stream>>>(hc, w_out, b_out, ctx);
}